// SegmentalEmotion_72773925863997
// MI455X (gfx1250) — compile-verified
//
#include <hip/hip_runtime.h>

typedef __attribute__((ext_vector_type(2))) float v2f;
typedef __attribute__((ext_vector_type(8))) float v8f;

// Replicates the host-side numpy plan: np.round is rint (round-half-to-even),
// computed in float64 under the default RNE rounding mode.
__device__ __forceinline__ void seg_plan(float len, int T,
                                         int& win, int& hop, int& n_eff,
                                         bool& fallback) {
    double dur = fmax((double)len, 0.001);
    double fps = (double)T / dur;
    long w = (long)__builtin_rint(fps * 1.0); if (w < 1) w = 1;   // WIN_SEC
    long h = (long)__builtin_rint(fps * 0.5); if (h < 1) h = 1;   // HOP_SEC
    long n = ((long)T >= w) ? ((long)T - w) / h + 1 : 0;
    fallback = (n == 0);
    long ne = (n > 1) ? n : 1;
    win = (int)w; hop = (int)h; n_eff = (int)ne;
}

// One block = (sample b, 16-segment group, 128 d-columns); 2 waves/block,
// each wave owns 4 d-tiles of 16 columns.
//   A[M][K] = weight(segment m0+M, time k0+K)  (1/count inside window, else 0)
//   B[K][N] = H[b, k0+K, dbase+N]
// accumulated with V_WMMA_F32_16X16X4_F32.
__global__ void seg_mean_wmma(const float* __restrict__ H,
                              const float* __restrict__ lens,
                              float* __restrict__ out,
                              int T, int maxS) {
    constexpr int D = 768;                      // fixed by the reference
    const int b     = blockIdx.x;
    const int m0    = blockIdx.y << 4;          // first segment of this group
    const int chunk = blockIdx.z;               // 128-column chunk of D
    const int lane  = threadIdx.x & 31;
    const int wave  = threadIdx.x >> 5;
    const int MN    = lane & 15;                // A-matrix M and B-matrix N
    const int dbase = chunk * 128 + wave * 64;  // 4 tiles of 16 columns

    int win, hop, n_eff; bool fallback;
    seg_plan(lens[b], T, win, hop, n_eff, fallback);

    // Per-lane segment parameters for segment s = m0 + MN.
    const int  sM     = m0 + MN;
    const bool validM = (sM < n_eff);
    int startM = fallback ? 0 : sM * hop;
    int endM   = fallback ? T : startM + win;
    if (startM > T) startM = T;
    if (endM   > T) endM   = T;
    int cnt = endM - startM; if (cnt < 1) cnt = 1;
    const float invM = validM ? (1.0f / (float)cnt) : 0.0f;  // 0 => padded slot stays 0

    v8f acc0 = {}, acc1 = {}, acc2 = {}, acc3 = {};

    if (m0 < n_eff) {  // group has >= 1 valid segment (uniform per block)
        // Union of this group's window spans: [kb, ke)
        int kb = fallback ? 0 : m0 * hop;
        int m1 = m0 + 15; if (m1 > n_eff - 1) m1 = n_eff - 1;
        int ke = fallback ? T : (m1 * hop + win);
        if (ke > T) ke = T;
        kb &= ~3;

        const int koff = (lane < 16) ? 0 : 2;   // K rows held by this half-wave
        const float* __restrict__ Hb = H + (size_t)b * (size_t)T * (size_t)D;
        const int off = koff * D + dbase + MN;  // loop-invariant lane offset

        // Main loop: k0+3 <= T-1 guaranteed -> no clamping, scalar base + fixed
        // vector offset (GVS addressing with immediate offsets).
        int ke_main = ke; if (ke_main > T - 3) ke_main = T - 3;
        const float* __restrict__ rbase = Hb + (size_t)kb * D;
        int k0 = kb;
        for (; k0 < ke_main; k0 += 4, rbase += 4 * (size_t)D) {
            const int ka = k0 + koff;
            const int kc = ka + 1;

            // A operand: VGPR0 = K{0,2}, VGPR1 = K{1,3} (ISA 16x4 f32 layout)
            v2f A;
            A.x = (ka >= startM && ka < endM) ? invM : 0.0f;
            A.y = (kc >= startM && kc < endM) ? invM : 0.0f;

            __builtin_prefetch(rbase + off + 8 * D, 0, 0);

            v2f B0; B0.x = rbase[off +  0]; B0.y = rbase[off + D +  0];
            v2f B1; B1.x = rbase[off + 16]; B1.y = rbase[off + D + 16];
            v2f B2; B2.x = rbase[off + 32]; B2.y = rbase[off + D + 32];
            v2f B3; B3.x = rbase[off + 48]; B3.y = rbase[off + D + 48];

            acc0 = __builtin_amdgcn_wmma_f32_16x16x4_f32(false, A, false, B0,
                                                         (short)0, acc0, false, false);
            acc1 = __builtin_amdgcn_wmma_f32_16x16x4_f32(false, A, false, B1,
                                                         (short)0, acc1, false, false);
            acc2 = __builtin_amdgcn_wmma_f32_16x16x4_f32(false, A, false, B2,
                                                         (short)0, acc2, false, false);
            acc3 = __builtin_amdgcn_wmma_f32_16x16x4_f32(false, A, false, B3,
                                                         (short)0, acc3, false, false);
        }

        // Clamped tail (at most one iteration; only when T is not 4-aligned
        // relative to ke). Weights are already 0 for k >= endM.
        for (; k0 < ke; k0 += 4) {
            const int ka = k0 + koff;
            const int kc = ka + 1;

            v2f A;
            A.x = (ka >= startM && ka < endM) ? invM : 0.0f;
            A.y = (kc >= startM && kc < endM) ? invM : 0.0f;

            int t0 = ka; if (t0 > T - 1) t0 = T - 1;
            int t1 = kc; if (t1 > T - 1) t1 = T - 1;
            const float* r0 = Hb + (size_t)t0 * D + dbase + MN;
            const float* r1 = Hb + (size_t)t1 * D + dbase + MN;

            v2f B0; B0.x = r0[ 0]; B0.y = r1[ 0];
            v2f B1; B1.x = r0[16]; B1.y = r1[16];
            v2f B2; B2.x = r0[32]; B2.y = r1[32];
            v2f B3; B3.x = r0[48]; B3.y = r1[48];

            acc0 = __builtin_amdgcn_wmma_f32_16x16x4_f32(false, A, false, B0,
                                                         (short)0, acc0, false, false);
            acc1 = __builtin_amdgcn_wmma_f32_16x16x4_f32(false, A, false, B1,
                                                         (short)0, acc1, false, false);
            acc2 = __builtin_amdgcn_wmma_f32_16x16x4_f32(false, A, false, B2,
                                                         (short)0, acc2, false, false);
            acc3 = __builtin_amdgcn_wmma_f32_16x16x4_f32(false, A, false, B3,
                                                         (short)0, acc3, false, false);
        }
    }

    // C/D layout: VGPR v holds row M = v (lanes 0-15) or v+8 (lanes 16-31),
    // N = lane&15.  Fast path: whole 16-row group in range (uniform branch).
    const int rowBase = (lane < 16) ? 0 : 8;
    float* obase = out + ((size_t)b * maxS + (m0 + rowBase)) * (size_t)D + dbase + MN;
    if (m0 + 16 <= maxS) {
#pragma unroll
        for (int v = 0; v < 8; ++v) {
            float* o = obase + (size_t)v * D;
            o[ 0] = acc0[v];
            o[16] = acc1[v];
            o[32] = acc2[v];
            o[48] = acc3[v];
        }
    } else {
#pragma unroll
        for (int v = 0; v < 8; ++v) {
            if (m0 + rowBase + v < maxS) {
                float* o = obase + (size_t)v * D;
                o[ 0] = acc0[v];
                o[16] = acc1[v];
                o[32] = acc2[v];
                o[48] = acc3[v];
            }
        }
    }
}

// Mask output: m[b, s] = (s < n_eff[b]) as 0/1 float, appended after S_pad.
__global__ void seg_mask(const float* __restrict__ lens,
                         float* __restrict__ mask, int T, int maxS) {
    const int b = blockIdx.x;
    int win, hop, n_eff; bool fallback;
    seg_plan(lens[b], T, win, hop, n_eff, fallback);
    for (int s = threadIdx.x; s < maxS; s += blockDim.x)
        mask[(size_t)b * maxS + s] = (s < n_eff) ? 1.0f : 0.0f;
}

extern "C" void kernel_launch(void* const* d_in, const int* in_sizes, int n_in,
                              void* d_out, int out_size, void* d_ws, size_t ws_size,
                              hipStream_t stream) {
    const float* H    = (const float*)d_in[0];
    const float* lens = (const float*)d_in[1];
    float* out        = (float*)d_out;

    const int B = in_sizes[1];                 // 32
    const int D = 768;                         // fixed by the reference
    const int T = in_sizes[0] / (B * D);       // 1500
    // out_size = B*maxS*D (S_pad) + B*maxS (mask)
    const int maxS = out_size / (B * (D + 1));

    const int nM     = (maxS + 15) / 16;
    const int nChunk = D / 128;                // 6 chunks of 128 columns

    dim3 grid(B, nM, nChunk);
    seg_mean_wmma<<<grid, 64, 0, stream>>>(H, lens, out, T, maxS);

    float* mask = out + (size_t)B * (size_t)maxS * (size_t)D;
    seg_mask<<<B, 64, 0, stream>>>(lens, mask, T, maxS);
}